// SubCenterArcFace_71683004170623
// MI455X (gfx1250) — compile-verified
//
#include <hip/hip_runtime.h>
#include <math.h>

#define NUM_CLASSES 50000
#define KSUB        3
#define EMBED_DIM   512
#define BATCH       1024
#define CK          (NUM_CLASSES * KSUB)   // 150000
#define MARGIN      0.5f
#define SCALEF      30.0f
#define EPSF        1e-7f

#define GROUPS      (CK / 48)              // 3125 groups; 48 subcenters = 16 classes each
#define WAVES       8
#define CHUNKS      ((GROUPS + WAVES - 1) / WAVES)   // 391
#define ROW_TILES   (BATCH / 16)           // 64
#define ASTRIDE     516                    // padded LDS row stride (floats): 16B-aligned, bank-conflict-free

typedef __attribute__((ext_vector_type(2))) float v2f;
typedef __attribute__((ext_vector_type(8))) float v8f;

// ---------------- kernel 1: L2-normalize embedding rows (1 wave per row) ----------------
__global__ __launch_bounds__(256) void normalize_rows_kernel(
    const float* __restrict__ x, float* __restrict__ y, int nrows) {
  const int wv = threadIdx.x >> 5, lane = threadIdx.x & 31;
  const int row = blockIdx.x * (blockDim.x >> 5) + wv;
  if (row >= nrows) return;
  const float4* xp = (const float4*)(x + (size_t)row * EMBED_DIM);
  float4 v[4];
  float ss = 0.f;
#pragma unroll
  for (int i = 0; i < 4; ++i) {
    v[i] = xp[lane + 32 * i];
    ss += v[i].x * v[i].x + v[i].y * v[i].y + v[i].z * v[i].z + v[i].w * v[i].w;
  }
#pragma unroll
  for (int off = 16; off; off >>= 1) ss += __shfl_xor(ss, off, 32);
  const float r = 1.0f / sqrtf(ss);
  float4* yp = (float4*)(y + (size_t)row * EMBED_DIM);
#pragma unroll
  for (int i = 0; i < 4; ++i) {
    float4 t = v[i];
    t.x *= r; t.y *= r; t.z *= r; t.w *= r;
    yp[lane + 32 * i] = t;
  }
}

// -------- kernel 2: fused WMMA GEMM + weight-norm + subcenter-max + margin + online LSE --------
// Grid = CHUNKS blocks. Each block owns 8 wave-groups of weight columns (768 KB, L2-resident)
// and loops over all 64 batch row-tiles, so weight leaves HBM approximately once.
__global__ __launch_bounds__(256) void arcface_gemm_kernel(
    const float* __restrict__ enorm, const float* __restrict__ weight,
    const int* __restrict__ labels,
    float* __restrict__ partial /* [BATCH][CHUNKS] as float2 */,
    float* __restrict__ tgt /* [BATCH] */) {
  __shared__ float As[16 * ASTRIDE];            // 33,024 B: 16 rows of normalized embeddings
  __shared__ float cosbuf[WAVES][16][49];       // 25,088 B: per-wave 16 rows x 48 subcenter cosines
  __shared__ float combM[WAVES][16];
  __shared__ float combS[WAVES][16];
  __shared__ int   lbl[16];

  const int tid  = threadIdx.x;
  const int wv   = tid >> 5;
  const int lane = tid & 31;
  const int g    = blockIdx.x * WAVES + wv;     // class-group handled by this wave
  const bool active = (g < GROUPS);             // wave-uniform => EXEC all-ones inside
  const int half = lane >> 4;                   // 0: K pair (k0,k0+1)   1: (k0+2,k0+3)
  const int sub  = lane & 15;                   // A: M row / B: weight (column) row
  const int k2   = 2 * half;
  const int base = g * 48;

  float wiv[3] = {0.f, 0.f, 0.f};               // inverse L2 norms of this lane's 3 weight rows

  for (int rt = 0; rt < ROW_TILES; ++rt) {
    const int rowBase = rt * 16;

    // cooperative load of the 16x512 A tile (float4, coalesced; L2-resident enorm)
    for (int i = tid; i < 16 * 128; i += 256) {
      const int r  = i >> 7;
      const int k4 = (i & 127) << 2;
      float4 v = *(const float4*)(enorm + (size_t)(rowBase + r) * EMBED_DIM + k4);
      *(float4*)(&As[r * ASTRIDE + k4]) = v;
      if (rt + 1 < ROW_TILES)                   // gfx1250 global_prefetch_b8 for next tile
        __builtin_prefetch(enorm + (size_t)(rowBase + 16 + r) * EMBED_DIM + k4, 0, 1);
    }
    if (tid < 16) lbl[tid] = labels[rowBase + tid];
    __syncthreads();                            // barrier 1: A tile + labels visible

    if (active) {
#pragma unroll
      for (int t = 0; t < 3; ++t) {
        const int wr = base + t * 16 + sub;
        const float* __restrict__ wp = weight + (size_t)wr * EMBED_DIM + k2;
        const float* ap = &As[sub * ASTRIDE + k2];
        v8f acc = {0.f, 0.f, 0.f, 0.f, 0.f, 0.f, 0.f, 0.f};
        if (rt == 0) {
          // first pass: also accumulate this lane's half of the weight-row sum of squares
          float bss = 0.f;
#pragma unroll 4
          for (int k0 = 0; k0 < EMBED_DIM; k0 += 4) {
            v2f a = *(const v2f*)(ap + k0);
            v2f b = *(const v2f*)(wp + k0);
            bss += b.x * b.x + b.y * b.y;
            acc = __builtin_amdgcn_wmma_f32_16x16x4_f32(
                false, a, false, b, (short)0, acc, false, false);
          }
          bss += __shfl_xor(bss, 16, 32);       // lanes l and l^16 hold complementary K halves
          wiv[t] = 1.0f / sqrtf(bss);
        } else {
#pragma unroll 4
          for (int k0 = 0; k0 < EMBED_DIM; k0 += 4) {
            v2f a = *(const v2f*)(ap + k0);     // ds_load_b64, conflict-free (stride 516)
            v2f b = *(const v2f*)(wp + k0);     // global float2, L2-resident after first tile
            acc = __builtin_amdgcn_wmma_f32_16x16x4_f32(
                false, a, false, b, (short)0, acc, false, false);
          }
        }
        const float w_t = wiv[t];
#pragma unroll
        for (int r = 0; r < 8; ++r)             // D layout: VGPR r -> rows r / r+8
          cosbuf[wv][r + 8 * half][t * 16 + sub] = acc[r] * w_t;
      }
    }
    __syncthreads();                            // barrier 2: cosines visible

    if (active) {
      // each lane owns (row = lane>>1, 8 of the 16 classes in this group)
      const int prow  = lane >> 1;
      const int jbase = (lane & 1) * 8;
      const int myLbl = lbl[prow];
      float m = -INFINITY, s = 0.f;
#pragma unroll
      for (int jj = 0; jj < 8; ++jj) {
        const int j  = jbase + jj;
        const int si = 3 * j;
        float c = fmaxf(cosbuf[wv][prow][si],
                  fmaxf(cosbuf[wv][prow][si + 1], cosbuf[wv][prow][si + 2]));
        c = fminf(fmaxf(c, -1.0f + EPSF), 1.0f - EPSF);
        const int cls = g * 16 + j;
        float v;
        if (cls == myLbl) {
          v = SCALEF * cosf(acosf(c) + MARGIN); // marginal logit for the target class
          tgt[rowBase + prow] = v;              // unique writer across the whole grid
        } else {
          v = SCALEF * c;
        }
        const float mn = fmaxf(m, v);
        s = s * expf(m - mn) + expf(v - mn);
        m = mn;
      }
      // merge the two lanes that share a row
      const float m2 = __shfl_xor(m, 1, 32);
      const float s2 = __shfl_xor(s, 1, 32);
      const float mn = fmaxf(m, m2);
      const float sc = s * expf(m - mn) + s2 * expf(m2 - mn);
      if ((lane & 1) == 0) { combM[wv][prow] = mn; combS[wv][prow] = sc; }
    } else if (lane < 16) {
      combM[wv][lane] = -INFINITY;
      combS[wv][lane] = 0.f;
    }
    __syncthreads();                            // barrier 3: per-wave partials visible

    if (tid < 16) {  // combine the 8 waves' partials for each row, emit one float2 per chunk
      float M = -INFINITY, S = 0.f;
#pragma unroll
      for (int w = 0; w < WAVES; ++w) {
        const float mw = combM[w][tid], sw = combS[w][tid];
        if (sw > 0.f) {
          const float mn = fmaxf(M, mw);
          S = S * expf(M - mn) + sw * expf(mw - mn);
          M = mn;
        }
      }
      ((float2*)partial)[(size_t)(rowBase + tid) * CHUNKS + blockIdx.x] = make_float2(M, S);
    }
    // no 4th barrier needed: next-iteration LDS writes are separated from this
    // iteration's readers by barriers 1/2 of the next iteration (per-buffer analysis)
  }
}

// ---------------- kernel 3: per-row log-sum-exp over chunks -> row loss ----------------
__global__ __launch_bounds__(256) void row_reduce_kernel(
    const float* __restrict__ partial, const float* __restrict__ tgt,
    float* __restrict__ rowloss) {
  __shared__ float sm[256], ss[256];
  const int row = blockIdx.x, tid = threadIdx.x;
  const float2* p = (const float2*)partial + (size_t)row * CHUNKS;
  float M = -INFINITY, S = 0.f;
  for (int c = tid; c < CHUNKS; c += 256) {     // CHUNKS=391 > 256: every thread gets >=1
    const float2 v = p[c];
    const float mn = fmaxf(M, v.x);
    S = S * expf(M - mn) + v.y * expf(v.x - mn);
    M = mn;
  }
  sm[tid] = M; ss[tid] = S;
  __syncthreads();
  for (int off = 128; off > 0; off >>= 1) {
    if (tid < off) {
      const float m2 = sm[tid + off], s2 = ss[tid + off];
      const float mn = fmaxf(sm[tid], m2);
      ss[tid] = ss[tid] * expf(sm[tid] - mn) + s2 * expf(m2 - mn);
      sm[tid] = mn;
    }
    __syncthreads();
  }
  if (tid == 0) rowloss[row] = (sm[0] + logf(ss[0])) - tgt[row];
}

// ---------------- kernel 4: mean over rows ----------------
__global__ __launch_bounds__(256) void final_mean_kernel(
    const float* __restrict__ rowloss, float* __restrict__ out) {
  __shared__ float sm[256];
  const int tid = threadIdx.x;
  float s = 0.f;
  for (int i = tid; i < BATCH; i += 256) s += rowloss[i];
  sm[tid] = s;
  __syncthreads();
  for (int off = 128; off > 0; off >>= 1) {
    if (tid < off) sm[tid] += sm[tid + off];
    __syncthreads();
  }
  if (tid == 0) out[0] = sm[0] / (float)BATCH;
}

extern "C" void kernel_launch(void* const* d_in, const int* in_sizes, int n_in,
                              void* d_out, int out_size, void* d_ws, size_t ws_size,
                              hipStream_t stream) {
  const float* emb    = (const float*)d_in[0];   // [1024, 512] f32
  const int*   labels = (const int*)  d_in[1];   // [1024] i32
  const float* weight = (const float*)d_in[2];   // [150000, 512] f32
  float* out = (float*)d_out;

  float* ws      = (float*)d_ws;                 // ~5.3 MB total
  float* enorm   = ws;                                   // 1024*512
  float* tgt     = enorm + (size_t)BATCH * EMBED_DIM;    // 1024
  float* rowloss = tgt + BATCH;                          // 1024
  float* partial = rowloss + BATCH;                      // 1024*391*2 (8B-aligned: even offset)

  normalize_rows_kernel<<<BATCH / 8, 256, 0, stream>>>(emb, enorm, BATCH);
  arcface_gemm_kernel<<<CHUNKS, 256, 0, stream>>>(enorm, weight, labels, partial, tgt);
  row_reduce_kernel<<<BATCH, 256, 0, stream>>>(partial, tgt, rowloss);
  final_mean_kernel<<<1, 256, 0, stream>>>(rowloss, out);
}